// SolidAngleOccp_components_25958782337251
// MI455X (gfx1250) — compile-verified
//
#include <hip/hip_runtime.h>
#include <hip/hip_bf16.h>

typedef __attribute__((ext_vector_type(2))) float v2f;
typedef __attribute__((ext_vector_type(8))) float v8f;

#define NPTS 4096
#define NFACE 2048
#define NCOMP 16
#define FACES_PER_BLOCK 128   // NFACE / gridDim.y(16)
#define WAVES_PER_BLOCK 8
#define PTS_PER_BLOCK (WAVES_PER_BLOCK * 16)

__device__ __forceinline__ float solid_angle_x2(float qx, float qy, float qz,
                                                const float* __restrict__ fv) {
    // v = vertex - query; a,b,c are the three vertices of the face
    float ax = fv[0] - qx, ay = fv[1] - qy, az = fv[2] - qz;
    float bx = fv[3] - qx, by = fv[4] - qy, bz = fv[5] - qz;
    float cx = fv[6] - qx, cy = fv[7] - qy, cz = fv[8] - qz;
    // det = a . (b x c)
    float crx = by * cz - bz * cy;
    float cry = bz * cx - bx * cz;
    float crz = bx * cy - by * cx;
    float det = ax * crx + ay * cry + az * crz;
    float na = sqrtf(ax * ax + ay * ay + az * az);
    float nb = sqrtf(bx * bx + by * by + bz * bz);
    float nc = sqrtf(cx * cx + cy * cy + cz * cz);
    float ab = ax * bx + ay * by + az * bz;
    float bc = bx * cx + by * cy + bz * cz;
    float ac = ax * cx + ay * cy + az * cz;
    float denom = na * nb * nc + bc * na + ac * nb + ab * nc;
    return atan2f(det, denom);
}

// Low 32 bits of a flat shared-aperture address == in-group LDS byte offset
// (ISA 10.2: LDS aperture occupies addr[63:32]).
__device__ __forceinline__ unsigned lds_offset(const void* p) {
    return (unsigned)(unsigned long long)p;
}

__global__ void zero_out_kernel(float* __restrict__ out, int n) {
    int i = blockIdx.x * blockDim.x + threadIdx.x;
    if (i < n) out[i] = 0.0f;
}

__global__ __launch_bounds__(256) void solid_angle_occ_kernel(
    const float* __restrict__ query,   // [NPTS][3]
    const float* __restrict__ face,    // [NFACE][3][3]
    const int* __restrict__ label,     // [NFACE]
    float* __restrict__ out)           // [NPTS][NCOMP]
{
    __shared__ __align__(16) float sface[FACES_PER_BLOCK * 9];   // 4608 B
    __shared__ __align__(16) int   slab[FACES_PER_BLOCK];        // 512 B

    const int tid = threadIdx.x;
    const int chunk = blockIdx.y;                    // face chunk
    const int fbase = chunk * FACES_PER_BLOCK;

    // ---- Async global -> LDS staging (CDNA5 GLOBAL_LOAD_ASYNC_TO_LDS path) ----
    // Faces: 4608 B = 288 x b128 transfers, spread across 256 lanes.
    {
        const float* gface = face + (size_t)fbase * 9;   // 16B-aligned chunk base
        const unsigned sbase = lds_offset(sface);
        for (int i = tid; i < (FACES_PER_BLOCK * 9) / 4; i += 256) {
            unsigned lds = sbase + (unsigned)(i * 16);
            int      off = i * 16;
            asm volatile("global_load_async_to_lds_b128 %0, %1, %2"
                         :: "v"(lds), "v"(off), "s"(gface)
                         : "memory");
        }
        // Labels: 512 B = 32 x b128 transfers.
        const int* glab = label + fbase;
        const unsigned lbase = lds_offset(slab);
        if (tid < FACES_PER_BLOCK / 4) {
            unsigned lds = lbase + (unsigned)(tid * 16);
            int      off = tid * 16;
            asm volatile("global_load_async_to_lds_b128 %0, %1, %2"
                         :: "v"(lds), "v"(off), "s"(glab)
                         : "memory");
        }
        asm volatile("s_wait_asynccnt 0" ::: "memory");
    }
    __syncthreads();

    const int lane = tid & 31;
    const int wave = tid >> 5;
    const int m  = lane & 15;     // point row within tile / component column
    const int hi = lane >> 4;     // K half: lanes 16-31 handle K=2,3

    const int tile  = blockIdx.x * WAVES_PER_BLOCK + wave;
    const int pbase = tile * 16;

    // This lane's query point (lanes 16-31 duplicate lanes 0-15 -> broadcast load)
    const float qx = query[(pbase + m) * 3 + 0];
    const float qy = query[(pbase + m) * 3 + 1];
    const float qz = query[(pbase + m) * 3 + 2];

    v8f acc = {};  // C/D: 16 points x 16 components, f32

    for (int k = 0; k < FACES_PER_BLOCK; k += 4) {
        const int f0 = k + 2 * hi;       // A-fragment K mapping per ISA 7.12.2
        const int f1 = f0 + 1;
        const float* fv0 = &sface[f0 * 9];
        const float* fv1 = &sface[f1 * 9];

        v2f A, Bm;
        A.x = solid_angle_x2(qx, qy, qz, fv0);
        A.y = solid_angle_x2(qx, qy, qz, fv1);
        // One-hot segment matrix: B[k][n] = (clamp(label[face k]) == n);
        // labels were staged raw, apply the reference's (-1 -> 0) clamp here.
        int l0 = slab[f0]; l0 = l0 < 0 ? 0 : l0;
        int l1 = slab[f1]; l1 = l1 < 0 ? 0 : l1;
        Bm.x = (l0 == m) ? 1.0f : 0.0f;
        Bm.y = (l1 == m) ? 1.0f : 0.0f;

        acc = __builtin_amdgcn_wmma_f32_16x16x4_f32(
            /*neg_a=*/false, A, /*neg_b=*/false, Bm,
            /*c_mod=*/(short)0, acc, /*reuse_a=*/false, /*reuse_b=*/false);
    }

    // D layout: VGPR j -> point row (j + 8*hi), component column (lane&15).
    const float inv2pi = 0.15915494309189535f;
#pragma unroll
    for (int j = 0; j < 8; ++j) {
        const int point = pbase + j + 8 * hi;
        atomicAdd(&out[point * NCOMP + m], acc[j] * inv2pi);
    }
}

extern "C" void kernel_launch(void* const* d_in, const int* in_sizes, int n_in,
                              void* d_out, int out_size, void* d_ws, size_t ws_size,
                              hipStream_t stream) {
    const float* query = (const float*)d_in[0];  // (1, 4096, 3) f32
    const float* face  = (const float*)d_in[1];  // (1, 2048, 3, 3) f32
    const int*   label = (const int*)d_in[2];    // (1, 2048) i32
    float* out = (float*)d_out;                  // (1, 4096, 16) f32

    zero_out_kernel<<<(out_size + 255) / 256, 256, 0, stream>>>(out, out_size);

    dim3 grid(NPTS / PTS_PER_BLOCK, NFACE / FACES_PER_BLOCK);  // (32, 16)
    solid_angle_occ_kernel<<<grid, 256, 0, stream>>>(query, face, label, out);
}